// DGBM_48017734369474
// MI455X (gfx1250) — compile-verified
//
#include <hip/hip_runtime.h>
#include <hip/hip_bf16.h>
#include <math.h>

// ---------------- problem constants ----------------
#define Bn      8
#define Cd      64
#define Hh      128
#define Wd      128
#define HW_     (Hh*Wd)           // 16384
#define NEXP_   6
#define HEADS_  8
#define EPSV    1e-5f
#define Sf      (Bn*Cd*HW_)       // 8388608 elements per (8,64,128,128) tensor

typedef float v2f __attribute__((ext_vector_type(2)));
typedef float v8f __attribute__((ext_vector_type(8)));

#if defined(__gfx1250__) && __has_builtin(__builtin_amdgcn_wmma_f32_16x16x4_f32)
#define HAS_WMMA_F32 1
#else
#define HAS_WMMA_F32 0
#endif

// D = A(16x4) * B(4x16) + C, fp32 WMMA (CDNA5).
// A: lane&15 = M row; lanes0-15 carry K={0,1} in {x,y}, lanes16-31 K={2,3}.
// B: lane&15 = N col; halves carry K rows the same way.
// C/D: v8f, elem i -> row i (lanes 0-15) / row i+8 (lanes 16-31), col = lane&15.
__device__ __forceinline__ v8f wmma_f32(v2f a, v2f b, v8f c) {
#if HAS_WMMA_F32
  return __builtin_amdgcn_wmma_f32_16x16x4_f32(false, a, false, b, (short)0, c,
                                               false, false);
#else
  c[0] += a.x * b.x;  // placeholder for non-gfx1250 parse passes
  return c;
#endif
}

// ---------------- elementwise / reductions ----------------
__global__ void k_fill(float* p, long n, float v) {
  long i = (long)blockIdx.x * blockDim.x + threadIdx.x;
  long st = (long)gridDim.x * blockDim.x;
  for (; i < n; i += st) p[i] = v;
}

__global__ void k_bn_stats(const float* __restrict__ x, float* __restrict__ mv) {
  int c = blockIdx.x, tid = threadIdx.x;
  float s = 0.f, s2 = 0.f;
  for (int i = tid; i < Bn * HW_; i += blockDim.x) {
    int b = i / HW_, p = i % HW_;
    float v = x[((long)b * Cd + c) * HW_ + p];
    s += v; s2 += v * v;
  }
  __shared__ float a1[256], a2[256];
  a1[tid] = s; a2[tid] = s2; __syncthreads();
  for (int k = 128; k > 0; k >>= 1) {
    if (tid < k) { a1[tid] += a1[tid + k]; a2[tid] += a2[tid + k]; }
    __syncthreads();
  }
  if (tid == 0) {
    float inv = 1.f / (float)(Bn * HW_);
    float m = a1[0] * inv;
    mv[c] = m;
    mv[64 + c] = a2[0] * inv - m * m;   // biased var (BN training mode)
  }
}

__global__ void k_bn_apply(const float* __restrict__ x, const float* __restrict__ mv,
                           const float* __restrict__ g, const float* __restrict__ bt,
                           float* __restrict__ xn) {
  long idx = (long)blockIdx.x * blockDim.x + threadIdx.x;
  if (idx >= Sf) return;
  int c = (int)((idx / HW_) % Cd);
  xn[idx] = (x[idx] - mv[c]) * rsqrtf(mv[64 + c] + EPSV) * g[c] + bt[c];
}

// ------------- WMMA implicit-GEMM conv (1x1 / 3x3, pad=KSZ/2), templated -------------
// One wave: 16 output channels x 64 pixels (4 N-sub-tiles sharing the A fragment).
// All compile-time: divisions fold, edge guards vanish for KSZ==1, loads are
// unconditional (clamped address + select) so EXEC stays all-ones for WMMA.
template <int INC, int OUTC, int KSZ, int ACT, bool RES>
__global__ void k_wmma_conv(const float* __restrict__ in, const float* __restrict__ w,
                            const float* __restrict__ bias, const float* __restrict__ res,
                            float* __restrict__ out) {
  constexpr int MT = (OUTC + 15) / 16;
  constexpr int TAPS = KSZ * KSZ;
  constexpr int PAD = KSZ / 2;
  constexpr int KSTEPS = (TAPS * INC) / 4;
  constexpr int NG = HW_ / 64;                  // 256 pixel-groups per image
  const int nwaves = Bn * MT * NG;
  int wvid = (int)((blockIdx.x * blockDim.x + threadIdx.x) >> 5);
  if (wvid >= nwaves) return;                   // wave-uniform
  int lane = threadIdx.x & 31;
  int b = wvid / (MT * NG);
  int r = wvid % (MT * NG);
  int mt = r / NG, g = r % NG;
  int y = g >> 1, x0 = (g & 1) << 6;            // 64 consecutive pixels of one row
  int col = lane & 15;
  bool hi = lane >= 16;
  int am = mt * 16 + col;                       // A row (output channel)
  const float* inb = in + (long)b * INC * HW_;
  v8f acc[4] = {};
#pragma unroll 4
  for (int kk = 0; kk < KSTEPS; ++kk) {
    const int kb = kk * 4;
    const int tap = kb / INC;                   // compile-time divisor
    const int ic = kb % INC + (hi ? 2 : 0);
    const int dy = tap / KSZ - PAD, dx = tap % KSZ - PAD;
    v2f a;
    const bool mok = (OUTC % 16 == 0) || (am < OUTC);
    const float* wp = w + ((long)am * INC + ic) * TAPS + tap;
    a.x = mok ? wp[0] : 0.f;
    a.y = mok ? wp[TAPS] : 0.f;                 // channel ic+1
    const int yy = y + dy;
    const bool oky = (KSZ == 1) || ((unsigned)yy < Hh);
    const float* ip = inb + (long)ic * HW_ + (oky ? yy : y) * Wd;
    __builtin_prefetch(ip + 2 * HW_, 0, 0);     // speculative: next channel pair
#pragma unroll
    for (int t = 0; t < 4; ++t) {
      const int xx = x0 + t * 16 + col + dx;
      const bool okx = (KSZ == 1) || ((unsigned)xx < Wd);
      const int xc = okx ? xx : 0;              // clamped: load always legal
      float v0 = ip[xc];
      float v1 = ip[xc + HW_];
      const bool ok = oky && okx;
      v2f bf;
      bf.x = ok ? v0 : 0.f;
      bf.y = ok ? v1 : 0.f;
      acc[t] = wmma_f32(a, bf, acc[t]);
    }
  }
#pragma unroll
  for (int t = 0; t < 4; ++t) {
    const int pbase = y * Wd + x0 + t * 16 + col;
#pragma unroll
    for (int i = 0; i < 8; ++i) {
      const int row = mt * 16 + i + (hi ? 8 : 0);
      if ((OUTC % 16 == 0) || (row < OUTC)) {
        float v = acc[t][i] + bias[row];
        const long o = ((long)b * OUTC + row) * HW_ + pbase;
        if (RES) v += res[o];
        if (ACT == 1) v = v > 0.f ? v : 0.f;
        else if (ACT == 2) v = 1.f / (1.f + __expf(-v));
        out[o] = v;
      }
    }
  }
}

// ------- depthwise 3x3 (pad=dil), templated: diff / relu / residual / MoE-accum -----
template <bool DIFF, bool RELU, bool RESADD, bool ACCUM>
__global__ void k_dw(const float* __restrict__ A, const float* __restrict__ Bm,
                     float sgn, const float* __restrict__ w,
                     const float* __restrict__ bias, const float* __restrict__ res,
                     const float* __restrict__ coef, int coefIdx,
                     float* __restrict__ out, int nchan, int dil, long n) {
  long idx = (long)blockIdx.x * blockDim.x + threadIdx.x;
  if (idx >= n) return;
  int p = (int)(idx % HW_);
  int c = (int)((idx / HW_) % nchan);
  int b = (int)(idx / ((long)HW_ * nchan));
  int y = p >> 7, x = p & 127;
  long base = ((long)b * nchan + c) * HW_;
  float wv[9];
#pragma unroll
  for (int t = 0; t < 9; ++t) wv[t] = w[c * 9 + t];
  float ts = 0.f;
#pragma unroll
  for (int t = 0; t < 9; ++t) {
    int yy = y + (t / 3 - 1) * dil;
    int xx = x + (t % 3 - 1) * dil;
    bool ok = (unsigned)yy < Hh && (unsigned)xx < Wd;
    long o = base + (ok ? yy : y) * Wd + (ok ? xx : x);  // clamped, always legal
    float v = A[o];
    if (DIFF) v -= Bm[o];
    ts += wv[t] * (ok ? v : 0.f);
  }
  float s = bias[c] + (DIFF ? sgn * ts : ts);
  if (RELU) s = s > 0.f ? s : 0.f;
  if (RESADD) s += res[idx];
  if (ACCUM) out[idx] += coef[b * NEXP_ + coefIdx] * s;
  else out[idx] = s;
}

// ---------------- attention: q/k row norms ----------------
__global__ void k_norms(const float* __restrict__ t1, float* __restrict__ norms) {
  int row = blockIdx.x, tid = threadIdx.x;       // 1024 rows: q(512) then k(512)
  int sel = row >> 9, r = row & 511;
  int b = r >> 6, cg = r & 63;
  const float* src = t1 + ((long)b * 192 + sel * 64 + cg) * HW_;
  float s = 0.f;
  for (int i = tid; i < HW_; i += blockDim.x) { float v = src[i]; s += v * v; }
  __shared__ float a1[256];
  a1[tid] = s; __syncthreads();
  for (int k = 128; k > 0; k >>= 1) { if (tid < k) a1[tid] += a1[tid + k]; __syncthreads(); }
  if (tid == 0) norms[row] = fmaxf(sqrtf(a1[0]), 1e-12f);
}

// gram[b,h,c,d] += sum_n q[c,n] k[d,n]  (WMMA over K-chunks, atomic fp32 combine)
#define NCHUNK 64
__global__ void k_gram(const float* __restrict__ t1, float* __restrict__ gram,
                       int nwaves) {
  int wvid = (int)((blockIdx.x * blockDim.x + threadIdx.x) >> 5);
  if (wvid >= nwaves) return;
  int lane = threadIdx.x & 31;
  int chunk = wvid % NCHUNK;
  int h = (wvid / NCHUNK) % HEADS_;
  int b = wvid / (NCHUNK * HEADS_);
  int n0 = chunk * (HW_ / NCHUNK);
  bool hi = lane >= 16;
  int mi = lane & 15;
  const float* q = t1 + ((long)b * 192 + h * 8) * HW_;
  const float* kp = q + 64 * HW_;
  const float* qr = q + (long)(mi & 7) * HW_;    // clamp row for safe loads
  const float* kr = kp + (long)(mi & 7) * HW_;
  bool mok = mi < 8;
  v8f acc = {};
#pragma unroll 4
  for (int kk = 0; kk < (HW_ / NCHUNK) / 4; ++kk) {
    int k0 = n0 + (kk << 2) + (hi ? 2 : 0);
    v2f a, bf;
    float q0 = qr[k0], q1 = qr[k0 + 1], kv0 = kr[k0], kv1 = kr[k0 + 1];
    a.x = mok ? q0 : 0.f;
    a.y = mok ? q1 : 0.f;
    bf.x = mok ? kv0 : 0.f;
    bf.y = mok ? kv1 : 0.f;
    acc = wmma_f32(a, bf, acc);
  }
#pragma unroll
  for (int i = 0; i < 8; ++i) {
    int cc = i + (hi ? 8 : 0);
    if (cc < 8 && mi < 8) {
      float* dst = gram + ((long)(b * HEADS_ + h) * 8 + cc) * 8 + mi;
      __hip_atomic_fetch_add(dst, acc[i], __ATOMIC_RELAXED, __HIP_MEMORY_SCOPE_AGENT);
    }
  }
}

// softmax over d with temp[h]/(|q_c||k_d|) folded in; in-place on gram
__global__ void k_softmax_attn(float* __restrict__ gram, const float* __restrict__ norms,
                               const float* __restrict__ temp) {
  int row = blockIdx.x * blockDim.x + threadIdx.x;
  if (row >= Bn * HEADS_ * 8) return;
  int cc = row & 7, h = (row >> 3) & 7, b = row >> 6;
  float nq = norms[b * 64 + h * 8 + cc];
  float l[8], mx = -INFINITY;
  for (int d = 0; d < 8; ++d) {
    float nk = norms[512 + b * 64 + h * 8 + d];
    l[d] = gram[row * 8 + d] * temp[h] / (nq * nk);
    mx = fmaxf(mx, l[d]);
  }
  float se = 0.f;
  for (int d = 0; d < 8; ++d) { l[d] = __expf(l[d] - mx); se += l[d]; }
  for (int d = 0; d < 8; ++d) gram[row * 8 + d] = l[d] / se;
}

// out[b, h*8+c, n] = sum_d a[c,d] v[d,n]   (M=8 padded to 16, K=8 -> 2 WMMA steps)
__global__ void k_av(const float* __restrict__ t1, const float* __restrict__ attnP,
                     float* __restrict__ out, int nwaves) {
  int wvid = (int)((blockIdx.x * blockDim.x + threadIdx.x) >> 5);
  if (wvid >= nwaves) return;
  int lane = threadIdx.x & 31;
  int nt = wvid % 1024;
  int h = (wvid / 1024) % HEADS_;
  int b = wvid / (1024 * HEADS_);
  int p0 = nt << 4, col = lane & 15;
  bool hi = lane >= 16;
  const float* v = t1 + ((long)b * 192 + 128 + h * 8) * HW_;
  const float* ap = attnP + (long)(b * HEADS_ + h) * 64;
  v8f acc = {};
#pragma unroll
  for (int kk = 0; kk < 2; ++kk) {
    int k0 = (kk << 2) + (hi ? 2 : 0);
    v2f a, bf;
    float a0 = ap[(col & 7) * 8 + k0], a1 = ap[(col & 7) * 8 + k0 + 1];
    a.x = col < 8 ? a0 : 0.f;
    a.y = col < 8 ? a1 : 0.f;
    bf.x = v[(long)k0 * HW_ + p0 + col];
    bf.y = v[(long)(k0 + 1) * HW_ + p0 + col];
    acc = wmma_f32(a, bf, acc);
  }
#pragma unroll
  for (int i = 0; i < 8; ++i) {
    int cc = i + (hi ? 8 : 0);
    if (cc < 8) out[((long)b * 64 + h * 8 + cc) * HW_ + p0 + col] = acc[i];
  }
}

// ---------------- pooling (max+mean over HW, or mean only), fused diff ----------
__global__ void k_pool(const float* __restrict__ A, const float* __restrict__ Bm,
                       float sgn, float* __restrict__ out, int mode) {
  int row = blockIdx.x, tid = threadIdx.x;      // row = b*64 + c
  long base = (long)row * HW_;
  float mx = -INFINITY, sm = 0.f;
  for (int i = tid; i < HW_; i += blockDim.x) {
    float v = A[base + i];
    if (Bm) v = sgn * (v - Bm[base + i]);
    mx = fmaxf(mx, v); sm += v;
  }
  __shared__ float smx[256], ssm[256];
  smx[tid] = mx; ssm[tid] = sm; __syncthreads();
  for (int k = 128; k > 0; k >>= 1) {
    if (tid < k) { smx[tid] = fmaxf(smx[tid], smx[tid + k]); ssm[tid] += ssm[tid + k]; }
    __syncthreads();
  }
  if (tid == 0) {
    float mean = ssm[0] / (float)HW_;
    out[row] = mode == 0 ? smx[0] + mean : mean;
  }
}

// noisy top-k gate -> per-(b,e) softmax coefficients
__global__ void k_gate(const float* __restrict__ p, const float* __restrict__ fc1w,
                       const float* __restrict__ fc1b, const float* __restrict__ fc0w,
                       const float* __restrict__ fc0b, float* __restrict__ cof) {
  int b = threadIdx.x;
  if (b >= Bn) return;
  float h[NEXP_], nz[NEXP_];
  for (int j = 0; j < NEXP_; ++j) {
    float a1 = fc1b[j], a0 = fc0b[j];
    for (int i = 0; i < Cd; ++i) {
      float pv = p[b * Cd + i];
      a1 += fc1w[j * Cd + i] * pv;
      a0 += fc0w[j * Cd + i] * pv;
    }
    h[j] = a1 > 0.f ? a1 : 0.2f * a1;                     // leaky_relu(0.2)
    nz[j] = a0 > 20.f ? a0 : log1pf(expf(a0));            // softplus
  }
  float mu = 0.f;
  for (int j = 0; j < NEXP_; ++j) mu += nz[j];
  mu /= NEXP_;
  float var = 0.f;
  for (int j = 0; j < NEXP_; ++j) { float d = nz[j] - mu; var += d * d; }
  float sd = sqrtf(var / (NEXP_ - 1));                    // ddof=1
  float s[NEXP_];
  for (int j = 0; j < NEXP_; ++j) s[j] = h[j] + (nz[j] - mu) / sd;
  bool sel[NEXP_] = {false, false, false, false, false, false};
  for (int t = 0; t < 3; ++t) {
    int bi = 0; float bv = -INFINITY;
    for (int j = 0; j < NEXP_; ++j)
      if (!sel[j] && s[j] > bv) { bv = s[j]; bi = j; }
    sel[bi] = true;
  }
  float mx = -INFINITY;
  for (int j = 0; j < NEXP_; ++j) if (sel[j]) mx = fmaxf(mx, h[j]);
  float se = 0.f, e[NEXP_];
  for (int j = 0; j < NEXP_; ++j) { e[j] = sel[j] ? expf(h[j] - mx) : 0.f; se += e[j]; }
  for (int j = 0; j < NEXP_; ++j) cof[b * NEXP_ + j] = e[j] / se;
}

// 64 -> relu 128 -> sigmoid 64, per batch row
__global__ void k_mlp(const float* __restrict__ pool, const float* __restrict__ w0,
                      const float* __restrict__ b0, const float* __restrict__ w1,
                      const float* __restrict__ b1, float* __restrict__ outv) {
  __shared__ float hs[128];
  int t = threadIdx.x;
  for (int b = 0; b < Bn; ++b) {
    float hv = b0[t];
    for (int i = 0; i < 64; ++i) hv += w0[t * 64 + i] * pool[b * 64 + i];
    hs[t] = hv > 0.f ? hv : 0.f;
    __syncthreads();
    if (t < 64) {
      float o = b1[t];
      for (int j = 0; j < 128; ++j) o += w1[t * 128 + j] * hs[j];
      outv[b * 64 + t] = 1.f / (1.f + expf(-o));
    }
    __syncthreads();
  }
}

// ecp = ec*ect + (1-ect)*amp[b,c]   (in place on ec)
__global__ void k_ecp(float* __restrict__ ec, const float* __restrict__ ect,
                      const float* __restrict__ amp) {
  long idx = (long)blockIdx.x * blockDim.x + threadIdx.x;
  if (idx >= Sf) return;
  int c = (int)((idx / HW_) % Cd);
  int b = (int)(idx / ((long)HW_ * Cd));
  float t = ect[idx];
  ec[idx] = ec[idx] * t + (1.f - t) * amp[b * Cd + c];
}

// out = dwX(ecp)*xb + dwY(ecp) + dwM(etp)*xh + dwN(etp)  (4 fused depthwise 3x3)
__global__ void k_final(const float* __restrict__ ecp, const float* __restrict__ etp,
                        const float* __restrict__ xb, const float* __restrict__ xh,
                        const float* __restrict__ wX, const float* __restrict__ bX,
                        const float* __restrict__ wY, const float* __restrict__ bY,
                        const float* __restrict__ wM, const float* __restrict__ bM,
                        const float* __restrict__ wN, const float* __restrict__ bN,
                        float* __restrict__ out) {
  long idx = (long)blockIdx.x * blockDim.x + threadIdx.x;
  if (idx >= Sf) return;
  int p = (int)(idx % HW_);
  int c = (int)((idx / HW_) % Cd);
  int b = (int)(idx / ((long)HW_ * Cd));
  int y = p >> 7, x = p & 127;
  long base = ((long)b * Cd + c) * HW_;
  float xw = bX[c], yw = bY[c], mw = bM[c], nw = bN[c];
#pragma unroll
  for (int t = 0; t < 9; ++t) {
    int yy = y + t / 3 - 1, xx = x + t % 3 - 1;
    bool ok = (unsigned)yy < Hh && (unsigned)xx < Wd;
    long o = base + (ok ? yy : y) * Wd + (ok ? xx : x);
    float e = ecp[o], tt = etp[o];
    e = ok ? e : 0.f; tt = ok ? tt : 0.f;
    xw += wX[c * 9 + t] * e;
    yw += wY[c * 9 + t] * e;
    mw += wM[c * 9 + t] * tt;
    nw += wN[c * 9 + t] * tt;
  }
  float r = xw * xb[idx] + yw + mw * xh[idx] + nw;
  __builtin_nontemporal_store(r, &out[idx]);   // written once, never re-read
}

// ---------------- host driver ----------------
static inline int eb(long n) { return (int)((n + 255) / 256); }

#define LAUNCH_CONV(INC, OUTC, KSZ, ACT, RES, in, w, bi, re, out)                    \
  k_wmma_conv<INC, OUTC, KSZ, ACT, RES>                                              \
      <<<(Bn * (((OUTC) + 15) / 16) * (HW_ / 64)) / 8, 256, 0, stream>>>(in, w, bi,  \
                                                                         re, out)

extern "C" void kernel_launch(void* const* d_in, const int* in_sizes, int n_in,
                              void* d_out, int out_size, void* d_ws, size_t ws_size,
                              hipStream_t stream) {
  (void)out_size; (void)ws_size;
  // locate x by its unique element count
  int xi = 0;
  for (int i = 0; i < n_in; ++i) if (in_sizes[i] == Sf) { xi = i; break; }
  const float* x = (const float*)d_in[xi];
  auto P = [&](int leaf) -> const float* {
    int idx = (xi == 0) ? leaf + 1 : leaf;   // params follow (or precede) x
    if (idx >= n_in) idx = n_in - 1;
    return (const float*)d_in[idx];
  };
  // jax pytree leaf order (sorted dict keys, tuples in order):
  const float *Mw = P(0), *Mb = P(1), *Nw = P(2), *Nb = P(3);
  const float *Xw = P(4), *Xbp = P(5), *Yw = P(6), *Yb = P(7);
  const float *dww = P(8), *dwb = P(9), *pjw = P(10), *pjb = P(11);
  const float *qkw = P(12), *qkb = P(13), *temp = P(14);
  const float *cvw = P(15), *cvb = P(16);
  const float *Cb1 = P(17), *Cb2 = P(18), *Cf0w = P(19), *Cf0b = P(20);
  const float *Cf1w = P(21), *Cf1b = P(22), *Cw1 = P(23), *Cw2 = P(24);
  const float *Tb1 = P(25), *Tb2 = P(26), *Tf0w = P(27), *Tf0b = P(28);
  const float *Tf1w = P(29), *Tf1b = P(30), *Tw1 = P(31), *Tw2 = P(32);
  const float *m0w = P(33), *m0b = P(34), *m1w = P(35), *m1b = P(36);
  const float *n0w = P(37), *n0b = P(38), *n1w = P(39), *n1b = P(40);
  const float *gam = P(41), *bet = P(42);
  const float *Uw[6], *Ub[6], *Vw[6], *Vb[6];
  for (int i = 0; i < 6; ++i) {
    Uw[i] = P(43 + 2 * i); Ub[i] = P(44 + 2 * i);
    Vw[i] = P(55 + 2 * i); Vb[i] = P(56 + 2 * i);
  }

  // workspace layout (floats)
  float* ws = (float*)d_ws;
  float* xn = ws;
  float* xb = ws + 1 * (size_t)Sf;
  float* xh = ws + 2 * (size_t)Sf;
  float* ec = ws + 3 * (size_t)Sf;
  float* et = ws + 4 * (size_t)Sf;
  float* t0 = ws + 5 * (size_t)Sf;   // 3S: qkv 1x1 out, then a@v out / ect,ett
  float* t1 = ws + 8 * (size_t)Sf;   // 3S: dw(qkv), then hdn / unet temps
  float* sm = ws + 11 * (size_t)Sf;  // small scalars
  float* bn_mv = sm;           // 128
  float* norms = sm + 128;     // 1024 (q:0-511, k:512-1023)
  float* gram  = sm + 1152;    // 4096
  float* pC    = sm + 5248;    // 512
  float* pT    = sm + 5760;    // 512
  float* cofC  = sm + 6272;    // 48
  float* cofT  = sm + 6320;    // 48
  float* pec   = sm + 6368;    // 512
  float* pet   = sm + 6880;    // 512
  float* eca   = sm + 7392;    // 512
  float* eta   = sm + 7904;    // 512
  float* u1 = t1;                       // (8,32,128,128)
  float* u2 = t1 + 4194304;             // (8,16,...)
  float* u3 = t1 + 6291456;             // (8, 8,...)
  float* u4 = t1 + 7340032;             // (8,16,...)
  float* u5 = t1 + 9437184;             // (8,32,...)
  float* ect = t0;
  float* ett = t0 + (size_t)Sf;

  // 1) batchnorm (training-mode batch stats)
  k_bn_stats<<<64, 256, 0, stream>>>(x, bn_mv);
  k_bn_apply<<<eb(Sf), 256, 0, stream>>>(x, bn_mv, gam, bet, xn);

  // 2) attention: qkv 1x1 (WMMA) -> dw 3x3 -> norms -> gram -> softmax -> a@v -> proj
  LAUNCH_CONV(64, 192, 1, 0, false, xn, qkw, qkb, nullptr, t0);
  k_dw<false, false, false, false><<<eb((long)Bn * 192 * HW_), 256, 0, stream>>>(
      t0, nullptr, 1.f, dww, dwb, nullptr, nullptr, 0, t1, 192, 1,
      (long)Bn * 192 * HW_);
  k_norms<<<1024, 256, 0, stream>>>(t1, norms);
  k_fill<<<16, 256, 0, stream>>>(gram, 4096, 0.f);
  int nw = Bn * HEADS_ * NCHUNK;
  k_gram<<<nw / 8, 256, 0, stream>>>(t1, gram, nw);
  k_softmax_attn<<<2, 256, 0, stream>>>(gram, norms, temp);
  nw = Bn * HEADS_ * 1024;
  k_av<<<nw / 8, 256, 0, stream>>>(t1, gram, t0, nw);
  LAUNCH_CONV(64, 64, 1, 0, true, t0, pjw, pjb, xn, xb);

  // 3) xh = xn + depthwise(xn)
  k_dw<false, false, true, false><<<eb(Sf), 256, 0, stream>>>(
      xn, nullptr, 1.f, cvw, cvb, xn, nullptr, 0, xh, 64, 1, (long)Sf);

  // 4) MoE gates (pooling fuses the xb-xh / xh-xb diff)
  k_pool<<<512, 256, 0, stream>>>(xb, xh, 1.f, pC, 0);
  k_gate<<<1, 32, 0, stream>>>(pC, Cf1w, Cf1b, Cf0w, Cf0b, cofC);
  k_pool<<<512, 256, 0, stream>>>(xb, xh, -1.f, pT, 0);
  k_gate<<<1, 32, 0, stream>>>(pT, Tf1w, Tf1b, Tf0w, Tf0b, cofT);

  // 5) experts: ec += cof * conv2(relu(conv1(diff)))   (dil 1 / dil 2)
  k_fill<<<8192, 256, 0, stream>>>(ec, 2L * Sf, 0.f);   // ec & et contiguous
  for (int e = 0; e < NEXP_; ++e) {
    k_dw<true, true, false, false><<<eb(Sf), 256, 0, stream>>>(
        xb, xh, 1.f, Cw1 + e * 576, Cb1 + e * 64, nullptr, nullptr, 0, t1, 64, 1,
        (long)Sf);
    k_dw<false, false, false, true><<<eb(Sf), 256, 0, stream>>>(
        t1, nullptr, 1.f, Cw2 + e * 576, Cb2 + e * 64, nullptr, cofC, e, ec, 64, 1,
        (long)Sf);
  }
  for (int e = 0; e < NEXP_; ++e) {
    k_dw<true, true, false, false><<<eb(Sf), 256, 0, stream>>>(
        xb, xh, -1.f, Tw1 + e * 576, Tb1 + e * 64, nullptr, nullptr, 0, t1, 64, 2,
        (long)Sf);
    k_dw<false, false, false, true><<<eb(Sf), 256, 0, stream>>>(
        t1, nullptr, 1.f, Tw2 + e * 576, Tb2 + e * 64, nullptr, cofT, e, et, 64, 2,
        (long)Sf);
  }

  // 6) channel attentions eca/eta from pooled means
  k_pool<<<512, 256, 0, stream>>>(ec, nullptr, 1.f, pec, 1);
  k_mlp<<<1, 128, 0, stream>>>(pec, m0w, m0b, m1w, m1b, eca);
  k_pool<<<512, 256, 0, stream>>>(et, nullptr, 1.f, pet, 1);
  k_mlp<<<1, 128, 0, stream>>>(pet, n0w, n0b, n1w, n1b, eta);

  // 7) U-Nets (all WMMA implicit GEMM): ect = unet(ec), ett = unet1(et)
  LAUNCH_CONV(64, 32, 1, 0, false, ec, Uw[0], Ub[0], nullptr, u1);
  LAUNCH_CONV(32, 16, 3, 1, false, u1, Uw[1], Ub[1], nullptr, u2);
  LAUNCH_CONV(16, 8, 3, 1, false, u2, Uw[2], Ub[2], nullptr, u3);
  LAUNCH_CONV(8, 16, 3, 1, false, u3, Uw[3], Ub[3], nullptr, u4);
  LAUNCH_CONV(16, 32, 3, 0, false, u4, Uw[4], Ub[4], nullptr, u5);
  LAUNCH_CONV(32, 64, 1, 2, false, u5, Uw[5], Ub[5], nullptr, ect);
  LAUNCH_CONV(64, 32, 1, 0, false, et, Vw[0], Vb[0], nullptr, u1);
  LAUNCH_CONV(32, 16, 3, 1, false, u1, Vw[1], Vb[1], nullptr, u2);
  LAUNCH_CONV(16, 8, 3, 1, false, u2, Vw[2], Vb[2], nullptr, u3);
  LAUNCH_CONV(8, 16, 3, 1, false, u3, Vw[3], Vb[3], nullptr, u4);
  LAUNCH_CONV(16, 32, 3, 0, false, u4, Vw[4], Vb[4], nullptr, u5);
  LAUNCH_CONV(32, 64, 1, 2, false, u5, Vw[5], Vb[5], nullptr, ett);

  // 8) ecp/etp in place, then fused final combine
  k_ecp<<<eb(Sf), 256, 0, stream>>>(ec, ect, eca);
  k_ecp<<<eb(Sf), 256, 0, stream>>>(et, ett, eta);
  k_final<<<eb(Sf), 256, 0, stream>>>(ec, et, xb, xh, Xw, Xbp, Yw, Yb, Mw, Mb, Nw, Nb,
                                      (float*)d_out);
}